// NeuronGemma4VisionPooler_74792560493244
// MI455X (gfx1250) — compile-verified
//
#include <hip/hip_runtime.h>

#define BB 8
#define SS 4096
#define HH 1152
#define LL 256

typedef __attribute__((ext_vector_type(2))) float v2f;
typedef __attribute__((ext_vector_type(8))) float v8f;

// ---------------------------------------------------------------------------
// Kernel 1: per-batch max_x+1  (reference: max(clamp(pos_x,0)) + 1)
// ---------------------------------------------------------------------------
__global__ void maxx_kernel(const int* __restrict__ pos, int* __restrict__ mx1) {
  __shared__ int red[256];
  const int b = blockIdx.x, t = threadIdx.x;
  int m = 0;
  for (int s = t; s < SS; s += 256) {
    int x = pos[((size_t)b * SS + s) * 2];
    m = max(m, max(x, 0));
  }
  red[t] = m;
  __syncthreads();
  for (int off = 128; off > 0; off >>= 1) {
    if (t < off) red[t] = max(red[t], red[t + off]);
    __syncthreads();
  }
  if (t == 0) mx1[b] = red[0] + 1;
}

// ---------------------------------------------------------------------------
// Kernel 2: kernel_idx[b,s] = (x//4) + (max_x1//4)*(y//4)
// ---------------------------------------------------------------------------
__global__ void idx_kernel(const int* __restrict__ pos, const int* __restrict__ mx1,
                           int* __restrict__ idxArr) {
  const int i = blockIdx.x * 256 + threadIdx.x;   // over B*S, exact multiple
  const int b = i / SS;
  const int x = max(pos[2 * i], 0);
  const int y = max(pos[2 * i + 1], 0);
  idxArr[i] = (x >> 2) + (mx1[b] >> 2) * (y >> 2);
}

// ---------------------------------------------------------------------------
// Kernel 3: sparse one-hot GEMM via V_WMMA_F32_16X16X4_F32
// grid: (l_tile=16, h_chunk=9, batch=8), block = 256 threads = 8 waves.
// Wave 0 compacts (ordered, race-free) the s-list for this l-tile; every wave
// then runs the K loop over compacted rows for its private 16-wide h slice.
// ---------------------------------------------------------------------------
__global__ void pool_kernel(const float* __restrict__ hs, const int* __restrict__ idxArr,
                            const unsigned char* __restrict__ padArr,
                            float* __restrict__ pooled, float* __restrict__ maskOut) {
  __shared__ unsigned list[SS];   // packed entries: [16:5]=s  [4:1]=m  [0]=not-padded
  __shared__ int flags[16];       // presence per local l (mask output)
  __shared__ int s_nchunks;

  const int lt = blockIdx.x;      // l-tile (16 l values)
  const int hc = blockIdx.y;      // h chunk of 128 columns
  const int b  = blockIdx.z;

  if (threadIdx.x < 32) {         // wave 0: ordered ballot compaction (wave32)
    const int lane = threadIdx.x;
    if (lane < 16) flags[lane] = 0;
    int cnt = 0;
    const int lbase = lt << 4;
    const int* ib = idxArr + (size_t)b * SS;
    const unsigned char* pb = padArr + (size_t)b * SS;
    for (int base = 0; base < SS; base += 32) {
      const int s = base + lane;
      const int m = ib[s] - lbase;
      const bool match = (unsigned)m < 16u;
      const unsigned mask = (unsigned)__ballot(match);
      if (match) {
        flags[m] = 1;                                 // presence regardless of padding
        const unsigned valid = pb[s] ? 0u : 1u;       // padded rows contribute weight 0
        const int p = cnt + __popc(mask & ((1u << lane) - 1u));
        list[p] = ((unsigned)s << 5) | ((unsigned)m << 1) | valid;
      }
      cnt += __popc(mask);
    }
    if (lane == 0) {
      const int padded = (cnt + 3) & ~3;              // pad to K-chunk of 4
      for (int i = cnt; i < padded; ++i) list[i] = 0u; // weight-0 filler
      s_nchunks = padded >> 2;
    }
  }
  __syncthreads();

  const int wave = threadIdx.x >> 5;
  const int lane = threadIdx.x & 31;
  const int half = lane >> 4;        // lane half selects K pair (0,1 vs 2,3) / M block
  const int n    = lane & 15;        // N column (== A's M row for this lane)
  const int h0   = hc * 128 + wave * 16;
  const float* hsb = hs + (size_t)b * SS * HH;

  v8f acc = {0.f, 0.f, 0.f, 0.f, 0.f, 0.f, 0.f, 0.f};
  const int nch = s_nchunks;
  for (int c = 0; c < nch; ++c) {
    // K layout guess per ISA A-matrix table: VGPR0 <-> K{0,2}, VGPR1 <-> K{1,3}
    const unsigned ex = list[4 * c + 2 * half];
    const unsigned ey = list[4 * c + 2 * half + 1];
    v2f a, bm;
    a.x = (((int)((ex >> 1) & 15u) == n) && (ex & 1u)) ? 0.0625f : 0.0f;
    a.y = (((int)((ey >> 1) & 15u) == n) && (ey & 1u)) ? 0.0625f : 0.0f;
    bm.x = hsb[(size_t)(ex >> 5) * HH + h0 + n];
    bm.y = hsb[(size_t)(ey >> 5) * HH + h0 + n];
    acc = __builtin_amdgcn_wmma_f32_16x16x4_f32(false, a, false, bm,
                                                (short)0, acc, false, false);
  }

  const float scale = 33.94112549695428f;   // sqrt(1152)
  float* op = pooled + ((size_t)b * LL + (size_t)(lt << 4)) * HH + h0 + n;
#pragma unroll
  for (int r = 0; r < 8; ++r) {             // D row M = r + 8*half (ISA C/D layout)
    op[(size_t)(r + 8 * half) * HH] = acc[r] * scale;
  }

  if (hc == 0 && threadIdx.x < 16) {        // mask once per (b, l)
    maskOut[(size_t)b * LL + (lt << 4) + threadIdx.x] = flags[threadIdx.x] ? 1.0f : 0.0f;
  }
}

// ---------------------------------------------------------------------------
extern "C" void kernel_launch(void* const* d_in, const int* in_sizes, int n_in,
                              void* d_out, int out_size, void* d_ws, size_t ws_size,
                              hipStream_t stream) {
  (void)in_sizes; (void)n_in; (void)out_size; (void)ws_size;
  const float*         hs  = (const float*)d_in[0];
  const int*           pos = (const int*)d_in[1];
  const unsigned char* pad = (const unsigned char*)d_in[2];  // all-false bools

  float* pooled = (float*)d_out;                       // [B, L, H] f32
  float* mask   = pooled + (size_t)BB * LL * HH;       // [B, L] as 0/1 floats

  int* mx1    = (int*)d_ws;                            // [B]
  int* idxArr = mx1 + BB;                              // [B*S]

  maxx_kernel<<<BB, 256, 0, stream>>>(pos, mx1);
  idx_kernel<<<(BB * SS) / 256, 256, 0, stream>>>(pos, mx1, idxArr);

  dim3 grid(LL / 16, HH / 128, BB);                    // 16 x 9 x 8
  pool_kernel<<<grid, 256, 0, stream>>>(hs, idxArr, pad, pooled, mask);
}